// CenterOfMass2DExtractor_73675868996056
// MI455X (gfx1250) — compile-verified
//
#include <hip/hip_runtime.h>
#include <hip/hip_bf16.h>

typedef __attribute__((ext_vector_type(2))) float v2f;
typedef __attribute__((ext_vector_type(8))) float v8f;

// Problem geometry (fixed by the reference): x is (8,4,256,256,64) f32.
#define NBF            32          // nb*nf slices
#define NZ             64
#define ROWS_TOTAL     65536       // 256*256 xy rows per slice
#define NBLK           64          // blocks per slice
#define ROWS_PER_BLK   1024        // ROWS_TOTAL / NBLK
#define HALF_ROWS      512
#define NPART          (NBLK * 2)  // partials (wave-halves) per slice

// ---------------------------------------------------------------------------
// Stage 1: streaming weighted reduction via V_WMMA_F32_16X16X4_F32.
// Block = 8 waves: wave = half(2) x ztile(4). Each wave reduces 512 xy-rows
// for one 16-wide z tile, chaining 128 WMMAs:  C += A(weights) x B(data).
//   A[0][k]=1, A[1][k]=j(row_k), A[2][k]=i(row_k), rows 3..15 = 0
//   => C row0 = mass, row1 = sum(x*j) (real), row2 = sum(x*i) (imag)
// Lane layout (ISA 7.12.2): lane L, vgpr v -> K = (L>>4)*2 + v;
//   A element = A[L&15][K], B element = X[rowbase+K][ztile*16 + (L&15)].
//
// Branch-free A generation: within a 256-row sub-block (64 iterations),
// i = row>>8 is constant and j = row&255 is affine (j = 4t + k), so the
// per-lane A element is an arithmetic progression: a += 4*wJ per iteration.
// ---------------------------------------------------------------------------
__global__ __launch_bounds__(256)
void com2d_partial_kernel(const float* __restrict__ x, float* __restrict__ ws) {
    const int blk   = blockIdx.x;        // 0..NBLK-1
    const int bf    = blockIdx.y;        // 0..NBF-1
    const int tid   = threadIdx.x;
    const int lane  = tid & 31;
    const int wave  = tid >> 5;
    const int half  = wave >> 2;         // row half
    const int ztile = wave & 3;          // 16-wide z tile
    const int m     = lane & 15;         // A-matrix row / B-matrix column
    const int kg    = lane >> 4;         // K-group: 0 -> K={0,1}, 1 -> K={2,3}
    const int k0    = kg << 1;

    // 0/1 lane-constant selectors for the three weighted sums.
    const float w1 = (m == 0) ? 1.0f : 0.0f;   // mass row
    const float wJ = (m == 1) ? 1.0f : 0.0f;   // column-index (real) row
    const float wI = (m == 2) ? 1.0f : 0.0f;   // row-index (imag) row
    const float aincr = wJ * 4.0f;             // j advances by 4 per WMMA

    const int R0 = blk * ROWS_PER_BLK + half * HALF_ROWS;  // multiple of 512
    const float* xbf = x + ((size_t)bf << 22);             // 65536*64 floats
    const float* p = xbf + (size_t)(R0 + k0) * NZ + ztile * 16 + m;

    v8f c = {0.f, 0.f, 0.f, 0.f, 0.f, 0.f, 0.f, 0.f};

    for (int sub = 0; sub < 2; ++sub) {      // i = row>>8 constant per sub
        const int rbase = R0 + sub * 256;
        const float iv = (float)(rbase >> 8);
        float a0 = w1 + wJ * (float)(k0)     + wI * iv;
        float a1 = w1 + wJ * (float)(k0 + 1) + wI * iv;
        #pragma unroll 4
        for (int t = 0; t < 64; ++t) {
            // Streaming loads: 512 MiB total >> 192 MB L2, mark non-temporal.
            float b0 = __builtin_nontemporal_load(p);
            float b1 = __builtin_nontemporal_load(p + NZ);
            if ((t & 3) == 0) {
                // 16 row-groups (16 KB) ahead; speculative, OOB-safe.
                __builtin_prefetch(p + 16 * 4 * NZ, 0, 0);
            }
            v2f A = {a0, a1};
            v2f B = {b0, b1};
            // 8 args: (neg_a, A, neg_b, B, c_mod, C, reuse_a, reuse_b)
            c = __builtin_amdgcn_wmma_f32_16x16x4_f32(
                    false, A, false, B, (short)0, c, false, false);
            a0 += aincr;
            a1 += aincr;
            p  += 4 * NZ;
        }
    }

    // C layout: VGPR v, lanes 0..15 hold row M=v, N=lane.
    // Rows 0/1/2 = mass / j-weighted / i-weighted sums for z = ztile*16+lane.
    if (lane < 16) {
        const size_t part = (size_t)(bf * NBLK + blk) * 2 + half;
        float* w = ws + part * (3 * NZ) + ztile * 16 + lane;
        w[0]      = c[0];   // mass
        w[NZ]     = c[1];   // real accumulator
        w[2 * NZ] = c[2];   // imag accumulator
    }
}

// ---------------------------------------------------------------------------
// Stage 2: deterministic fixed-order reduction of the 128 partials per
// (bf, z), then divide and emit interleaved complex64.
// ---------------------------------------------------------------------------
__global__ __launch_bounds__(256)
void com2d_finalize_kernel(const float* __restrict__ ws, float* __restrict__ out) {
    const int t = blockIdx.x * blockDim.x + threadIdx.x;
    if (t >= NBF * NZ) return;
    const int bf = t >> 6;
    const int z  = t & 63;
    const float* base = ws + (size_t)bf * NPART * (3 * NZ) + z;
    float s0 = 0.f, s1 = 0.f, s2 = 0.f;
    for (int pnum = 0; pnum < NPART; ++pnum) {
        const float* q = base + (size_t)pnum * (3 * NZ);
        s0 += q[0];
        s1 += q[NZ];
        s2 += q[2 * NZ];
    }
    out[2 * t]     = s1 / s0;   // real: column-index (j) weighted
    out[2 * t + 1] = s2 / s0;   // imag: row-index (i) weighted
}

extern "C" void kernel_launch(void* const* d_in, const int* in_sizes, int n_in,
                              void* d_out, int out_size, void* d_ws, size_t ws_size,
                              hipStream_t stream) {
    const float* x  = (const float*)d_in[0];
    float* ws       = (float*)d_ws;    // uses 32*128*192*4 = 3 MB
    float* out      = (float*)d_out;   // 2048 complex64 = 4096 floats

    dim3 grid(NBLK, NBF);
    com2d_partial_kernel<<<grid, 256, 0, stream>>>(x, ws);
    com2d_finalize_kernel<<<(NBF * NZ + 255) / 256, 256, 0, stream>>>(ws, out);
}